// LREN_78847009620091
// MI455X (gfx1250) — compile-verified
//
#include <hip/hip_runtime.h>

typedef __attribute__((ext_vector_type(2))) float v2f;
typedef __attribute__((ext_vector_type(8))) float v8f;

#define TT 64
#define DD 64
#define HH 256
#define LL 64
#define KK 128

__device__ __forceinline__ v8f wmma_f32(v2f a, v2f b, v8f c) {
  // V_WMMA_F32_16X16X4_F32 : D = A(16x4) x B(4x16) + C(16x16)
  return __builtin_amdgcn_wmma_f32_16x16x4_f32(
      /*neg_a=*/false, a, /*neg_b=*/false, b,
      /*c_mod=*/(short)0, c, /*reuse_a=*/false, /*reuse_b=*/false);
}

// ---------------------------------------------------------------------------
// Generic small transpose: out[c*R + r] = in[r*C + c]
// ---------------------------------------------------------------------------
__global__ void transpose_mat(const float* __restrict__ in, float* __restrict__ out,
                              int R, int C) {
  int i = blockIdx.x * blockDim.x + threadIdx.x;
  if (i < R * C) {
    int r = i / C, c = i % C;
    out[c * R + r] = in[i];
  }
}

// ---------------------------------------------------------------------------
// Transposed matrix powers of ko_W: PT[t][c][k] = (ko_W^t)[k][c], t = 0..63.
// ---------------------------------------------------------------------------
__global__ void pow_init(const float* __restrict__ koW, float* __restrict__ PT) {
  int i = blockIdx.x * blockDim.x + threadIdx.x;   // i = c*128 + k
  int c = i >> 7, k = i & 127;
  PT[i] = (k == c) ? 1.0f : 0.0f;                  // (ko_W^0)^T = I
  PT[KK * KK + i] = koW[k * KK + c];               // (ko_W^1)^T
}

// PT[first + bx] = T(P[a] @ P[b]) with a = n>>1, b = n - a (both computed).
// All operands stored transposed: X[r][k] == XT[k*128 + r].
__global__ void pow_step(float* __restrict__ PT, int first) {
  const int n = first + blockIdx.x;
  const int a = n >> 1, b = n - a;
  const float* AT = PT + (size_t)a * (KK * KK);
  const float* BT = PT + (size_t)b * (KK * KK);
  float* DT       = PT + (size_t)n * (KK * KK);

  __shared__ float As[16][KK + 2];                 // padded: bank-friendly
  const int row0 = blockIdx.y * 16;
  for (int i = threadIdx.x; i < 16 * KK; i += 256) {
    int k = i >> 4, r = i & 15;                    // coalesced 64B chunks
    As[r][k] = AT[k * KK + row0 + r];
  }
  __syncthreads();

  #pragma unroll
  for (int j = 0; j < 8; ++j) {
    int idx = threadIdx.x + j * 256;               // 2048 outputs per block
    int r = idx & 15, c = idx >> 4;                // 16 threads share c (broadcast)
    float s = 0.0f;
    #pragma unroll 8
    for (int k = 0; k < KK; ++k) s = fmaf(As[r][k], BT[c * KK + k], s);
    DT[c * KK + row0 + r] = s;
  }
}

// ---------------------------------------------------------------------------
// Fused MLP: enc_gt[:, :, 0:64] = x ; enc_gt[:, :, 64:128] = MLP(x)
// 2 waves / block, each wave owns 16 rows. Weights consumed transposed.
// ---------------------------------------------------------------------------
__global__ void __launch_bounds__(64) mlp_kernel(
    const float* __restrict__ x,   const float* __restrict__ W0t,
    const float* __restrict__ b0,  const float* __restrict__ W1t,
    const float* __restrict__ b1,  const float* __restrict__ W2t,
    float* __restrict__ enc_gt) {
  __shared__ float h1s[2][16 * 260];   // per-wave h1 tile, padded stride 260
  __shared__ float scrs[2][16 * 18];   // per-wave 16x16 relay tile, stride 18

  const int tid = threadIdx.x;
  const int w = tid >> 5, lane = tid & 31;
  const int m = lane & 15, kh = lane >> 4;
  const long blockRow = (long)blockIdx.x * 32;

  // ---- copy x tile into enc_gt[:, :, 0:64] (coalesced, whole block) ----
  for (int i = tid; i < 32 * 64; i += 64) {
    long r = blockRow + (i >> 6);
    enc_gt[r * KK + (i & 63)] = x[r * DD + (i & 63)];
  }

  float* h1  = h1s[w];
  float* scr = scrs[w];
  const long row0 = blockRow + w * 16;

  // ---- A-fragments of x (16x64), kept in registers ----
  v2f a1[16];
  const float* xr = x + (row0 + m) * DD;
  #pragma unroll
  for (int k4 = 0; k4 < 16; ++k4)
    a1[k4] = *reinterpret_cast<const v2f*>(xr + k4 * 4 + kh * 2);

  // ---- GEMM1: h1 = relu(x @ W0 + b0) -> LDS (C-layout -> A-layout relay) ----
  #pragma unroll 1
  for (int n = 0; n < 16; ++n) {
    v8f acc = {0, 0, 0, 0, 0, 0, 0, 0};
    const int col = n * 16 + m;
    const float* wcol = W0t + (size_t)col * DD;    // W0^T row = W0 column
    #pragma unroll
    for (int k4 = 0; k4 < 16; ++k4) {
      v2f bf = *reinterpret_cast<const v2f*>(wcol + k4 * 4 + kh * 2);
      acc = wmma_f32(a1[k4], bf, acc);
    }
    const float bias = b0[col];
    #pragma unroll
    for (int r = 0; r < 8; ++r)
      h1[(r + 8 * kh) * 260 + col] = fmaxf(acc[r] + bias, 0.0f);
  }

  // ---- reload h1 once as GEMM2 A-fragments, held in registers ----
  v2f a2[64];
  #pragma unroll
  for (int k4 = 0; k4 < 64; ++k4)
    a2[k4] = *reinterpret_cast<const v2f*>(h1 + m * 260 + k4 * 4 + kh * 2);

  // ---- GEMM2 tile-by-tile, each relu'd tile immediately feeds GEMM3 ----
  v8f zacc[4];
  #pragma unroll
  for (int zn = 0; zn < 4; ++zn) zacc[zn] = (v8f){0, 0, 0, 0, 0, 0, 0, 0};

  #pragma unroll 1
  for (int n2 = 0; n2 < 16; ++n2) {
    v8f acc = {0, 0, 0, 0, 0, 0, 0, 0};
    const int col = n2 * 16 + m;
    const float* wcol = W1t + (size_t)col * HH;    // W1^T row = W1 column
    #pragma unroll
    for (int k4 = 0; k4 < 64; ++k4) {              // K = 256
      v2f bf = *reinterpret_cast<const v2f*>(wcol + k4 * 4 + kh * 2);
      acc = wmma_f32(a2[k4], bf, acc);
    }
    const float bias = b1[col];
    #pragma unroll
    for (int r = 0; r < 8; ++r)
      scr[(r + 8 * kh) * 18 + m] = fmaxf(acc[r] + bias, 0.0f);  // C-layout store

    // consume scr as A-fragments for GEMM3 (K-chunk n2*16 .. n2*16+15)
    #pragma unroll
    for (int kk4 = 0; kk4 < 4; ++kk4) {
      const int kk = kk4 * 4 + kh * 2;
      v2f af2 = *reinterpret_cast<const v2f*>(scr + m * 18 + kk);
      #pragma unroll
      for (int zn = 0; zn < 4; ++zn) {
        const int zcol = zn * 16 + m;
        v2f bf2 = *reinterpret_cast<const v2f*>(W2t + (size_t)zcol * HH + n2 * 16 + kk);
        zacc[zn] = wmma_f32(af2, bf2, zacc[zn]);
      }
    }
  }

  // ---- write z into enc_gt[:, :, 64:128] ----
  #pragma unroll
  for (int zn = 0; zn < 4; ++zn) {
    #pragma unroll
    for (int r = 0; r < 8; ++r) {
      long row = row0 + r + 8 * kh;
      enc_gt[row * KK + 64 + zn * 16 + m] = zacc[zn][r];
    }
  }
}

// ---------------------------------------------------------------------------
// enc_traj[b, t, :] = z0[b, :] @ P[t];  one wave per (16-row b-tile, t).
// P consumed transposed (PT[t][c][k]) => single b64 B-fragment loads.
// ---------------------------------------------------------------------------
__global__ void __launch_bounds__(64) traj_kernel(
    const float* __restrict__ enc_gt, const float* __restrict__ PT,
    float* __restrict__ enc_traj) {
  const int tid = threadIdx.x;
  const int w = tid >> 5, lane = tid & 31;
  const int m = lane & 15, kh = lane >> 4;
  const int g = blockIdx.x * 2 + w;
  const int btile = g >> 6;            // 64 t's per b-tile
  const int t = g & 63;
  const long b0r = (long)btile * 16;

  // z0 A-fragments (16 rows x K=128), loaded once, reused for all 8 N-tiles
  v2f a[32];
  const float* z0 = enc_gt + (b0r + m) * (long)(TT * KK);  // row (b, t=0)
  #pragma unroll
  for (int k4 = 0; k4 < 32; ++k4)
    a[k4] = *reinterpret_cast<const v2f*>(z0 + k4 * 4 + kh * 2);

  const float* Mt = PT + (long)t * (KK * KK);
  #pragma unroll 1
  for (int n = 0; n < 8; ++n) {
    v8f acc = {0, 0, 0, 0, 0, 0, 0, 0};
    const int col = n * 16 + m;
    const float* mcol = Mt + (size_t)col * KK;
    #pragma unroll
    for (int k4 = 0; k4 < 32; ++k4) {
      v2f bf = *reinterpret_cast<const v2f*>(mcol + k4 * 4 + kh * 2);
      acc = wmma_f32(a[k4], bf, acc);
    }
    #pragma unroll
    for (int r = 0; r < 8; ++r) {
      long row = b0r + r + 8 * kh;
      enc_traj[(row * TT + t) * (long)KK + col] = acc[r];
    }
  }
}

// ---------------------------------------------------------------------------
extern "C" void kernel_launch(void* const* d_in, const int* in_sizes, int n_in,
                              void* d_out, int out_size, void* d_ws, size_t ws_size,
                              hipStream_t stream) {
  const float* x   = (const float*)d_in[0];
  const float* W0  = (const float*)d_in[1];
  const float* b0  = (const float*)d_in[2];
  const float* W1  = (const float*)d_in[3];
  const float* b1  = (const float*)d_in[4];
  const float* W2  = (const float*)d_in[5];
  const float* koW = (const float*)d_in[6];
  // d_in[7] = n_shifts (== 64, fixed by the problem shapes)

  const long Bsz = (long)in_sizes[0] / (TT * DD);          // 8192
  float* enc_gt   = (float*)d_out;
  float* enc_traj = enc_gt + Bsz * TT * KK;

  // workspace layout (floats): PT[64][128][128], then W0t, W1t, W2t
  float* PT  = (float*)d_ws;                               // 1,048,576 floats
  float* W0t = PT + (size_t)TT * KK * KK;                  // 64*256   = 16384
  float* W1t = W0t + DD * HH;                              // 256*256  = 65536
  float* W2t = W1t + HH * HH;                              // 256*64   = 16384

  // 0) transpose the weights (tiny one-shot kernels)
  transpose_mat<<<dim3((DD * HH + 255) / 256), 256, 0, stream>>>(W0, W0t, DD, HH);
  transpose_mat<<<dim3((HH * HH + 255) / 256), 256, 0, stream>>>(W1, W1t, HH, HH);
  transpose_mat<<<dim3((HH * LL + 255) / 256), 256, 0, stream>>>(W2, W2t, HH, LL);

  // 1) transposed matrix powers via log-doubling (7 tiny launches)
  pow_init<<<dim3((KK * KK) / 256), 256, 0, stream>>>(koW, PT);
  pow_step<<<dim3(1, 8),  256, 0, stream>>>(PT, 2);
  pow_step<<<dim3(2, 8),  256, 0, stream>>>(PT, 3);
  pow_step<<<dim3(4, 8),  256, 0, stream>>>(PT, 5);
  pow_step<<<dim3(8, 8),  256, 0, stream>>>(PT, 9);
  pow_step<<<dim3(16, 8), 256, 0, stream>>>(PT, 17);
  pow_step<<<dim3(31, 8), 256, 0, stream>>>(PT, 33);

  // 2) fused MLP -> enc_gt
  const long rows = Bsz * TT;                              // 524288
  mlp_kernel<<<dim3((unsigned)(rows / 32)), 64, 0, stream>>>(
      x, W0t, b0, W1t, b1, W2t, enc_gt);

  // 3) batched z0 @ P[t] -> enc_traj
  traj_kernel<<<dim3((unsigned)((Bsz / 16) * TT / 2)), 64, 0, stream>>>(
      enc_gt, PT, enc_traj);
}